// MyStrategicModel_35691178229867
// MI455X (gfx1250) — compile-verified
//
#include <hip/hip_runtime.h>

// Strategic-classification CCP fixed point, fully register-resident.
// One thread = one sample; 1300 serial projected-gradient steps per thread.
// Compute-bound fp32 VALU kernel: packed float2 state -> v_pk_fma_f32 /
// VOPD dual-issue on gfx1250; uniform params in SGPRs; raw v_rsq_f32.

typedef float v2f __attribute__((ext_vector_type(2)));

#define DEV __device__ __forceinline__

// fma(a.x,b.x, fma(a.y,b.y, c)) : dot product with folded additive constant.
DEV float dot2p(v2f a, v2f b, float c) {
  return fmaf(a.x, b.x, fmaf(a.y, b.y, c));
}

DEV v2f clip10(v2f x) {
  const v2f lo = {-10.0f, -10.0f};
  const v2f hi = { 10.0f,  10.0f};
  return __builtin_elementwise_min(__builtin_elementwise_max(x, lo), hi);
}

__global__ __launch_bounds__(256) void strat_ccp_kernel(
    const float* __restrict__ Xp, const float* __restrict__ wp,
    const float* __restrict__ bp, const float* __restrict__ vp,
    float* __restrict__ out, int B) {
  int i = blockIdx.x * blockDim.x + threadIdx.x;
  if (i >= B) return;

  // Uniform scalars -> SGPRs.
  v2f w = {wp[0], wp[1]};
  v2f v = {vp[0], vp[1]};
  const float b   = bp[0];
  const float bm1 = b - 1.0f;            // for g'(x):  s = x.w + b - 1
  const float bp1 = b + 1.0f;            // for f'(xt): sp = xt.w + b + 1
  const v2f  wh   = {0.5f * w.x, 0.5f * w.y};   // folds the 0.5 of g'/f' scale
  const v2f  pv   = {0.475f * v.x, 0.475f * v.y}; // (1-EPS)*COST * v
  const v2f  c95  = {0.95f, 0.95f};             // 1 - 2*EPS*COST (LR=1)

  // One global_load_b64 per sample.
  const v2f r = ((const v2f*)Xp)[i];

  // Per-sample loop invariants of the inner solve:
  //   x <- clip( 0.95*x + (fd + 0.05*r) - (s*rsq(s^2+1))*wh - [dv>0]*pv )
  const v2f  x0    = clip10(r);
  const float nrv  = -dot2p(r, v, 0.0f);        // -(r.v), folded into dv fma
  const v2f  rbase = {0.05f * r.x, 0.05f * r.y}; // 2*EPS*COST * r

  v2f xt = r;  // CCP linearization point starts at X (raw, unclipped)

  // 11 CCP rounds of 100 steps, then the final differentiable solve (200).
  for (int round = 0; round < 12; ++round) {
    // fd = (sp*rsq(sp^2+1)) * wh,  sp = xt.w + b + 1
    float sp  = dot2p(xt, w, bp1);
    float fm  = sp * __builtin_amdgcn_rsqf(fmaf(sp, sp, 1.0f));
    v2f fmv   = {fm, fm};
    v2f base  = __builtin_elementwise_fma(fmv, wh, rbase); // fd + 0.05*r
    v2f basem = base - pv;  // addend when the hinge [d.v > 0] is active

    const int steps = (round == 11) ? 200 : 100;
    v2f x = x0;  // each inner solve restarts from clip(X)
#pragma unroll 4
    for (int k = 0; k < steps; ++k) {
      float s   = dot2p(x, w, bm1);                                // 2 fma
      float m   = s * __builtin_amdgcn_rsqf(fmaf(s, s, 1.0f));     // fma+rsq+mul
      float dv  = dot2p(x, v, nrv);                                // 2 fma
      v2f bsel  = (dv > 0.0f) ? basem : base;                      // cmp + 2 cndmask
      v2f nm    = {-m, -m};                                        // neg folds into fma
      v2f acc   = __builtin_elementwise_fma(nm, wh, bsel);         // pk_fma
      x         = __builtin_elementwise_fma(c95, x, acc);          // pk_fma
      x         = clip10(x);                                       // dual max :: min
    }
    xt = x;
  }

  out[i] = dot2p(xt, w, b);

}

extern "C" void kernel_launch(void* const* d_in, const int* in_sizes, int n_in,
                              void* d_out, int out_size, void* d_ws, size_t ws_size,
                              hipStream_t stream) {
  const float* X = (const float*)d_in[0];
  const float* w = (const float*)d_in[1];
  const float* b = (const float*)d_in[2];
  const float* v = (const float*)d_in[3];
  float* out = (float*)d_out;

  const int B = in_sizes[0] / 2;  // X is [B, 2]
  const int threads = 256;        // 8 wave32 waves per workgroup
  const int blocks = (B + threads - 1) / threads;
  strat_ccp_kernel<<<blocks, threads, 0, stream>>>(X, w, b, v, out, B);
}